// CroquetGNN_43215960932690
// MI455X (gfx1250) — compile-verified
//
#include <hip/hip_runtime.h>
#include <math.h>

typedef __attribute__((ext_vector_type(2))) float v2f;
typedef __attribute__((ext_vector_type(8))) float v8f;

#define ATOMIC_ADD_F32(p, v) \
  __hip_atomic_fetch_add((p), (v), __ATOMIC_RELAXED, __HIP_MEMORY_SCOPE_AGENT)

// ---------------------------------------------------------------------------
// 1) Per-call init: deg[i]=1 (self-loop), acc1=0, acc2=0
// ---------------------------------------------------------------------------
__global__ void gcn_init(float* __restrict__ deg, float* __restrict__ acc1,
                         float* __restrict__ acc2, int N) {
  int i = blockIdx.x * blockDim.x + threadIdx.x;
  if (i < N * 16) acc1[i] = 0.0f;
  if (i < N) {
    deg[i]  = 1.0f;   // self-loop contribution to degree
    acc2[i] = 0.0f;
  }
}

// ---------------------------------------------------------------------------
// 2) deg[dst] += 1 over the E listed edges
// ---------------------------------------------------------------------------
__global__ void gcn_degree(const int* __restrict__ dst, float* __restrict__ deg,
                           int E) {
  int e = blockIdx.x * blockDim.x + threadIdx.x;
  if (e >= E) return;
  ATOMIC_ADD_F32(&deg[dst[e]], 1.0f);
}

// ---------------------------------------------------------------------------
// 3) dinv = rsqrt(deg)   (deg >= 1 always, so no zero guard needed)
// ---------------------------------------------------------------------------
__global__ void gcn_dinv(float* __restrict__ deg, int N) {
  int i = blockIdx.x * blockDim.x + threadIdx.x;
  if (i >= N) return;
  deg[i] = rsqrtf(deg[i]);
}

// ---------------------------------------------------------------------------
// 4) h1 = x @ W1 via V_WMMA_F32_16X16X4_F32. One wave handles 16 nodes.
//    A (16x4, M x K): lanes 0-15 carry K=0,1 in a.x/a.y; lanes 16-31 carry
//    K=2,3 (K=3 is the zero pad since F_in=3).
//    B (4x16, K x N): mirrored layout, lane index = output feature n.
//    C/D (16x16 f32): VGPR r -> row (r + 8*half), col = lane & 15.
//    Loads are unconditional; zero-pad is a value select (v_cndmask, no exec
//    branches). Stores take a wave-uniform fast path for full tiles.
// ---------------------------------------------------------------------------
__global__ void __launch_bounds__(256)
gcn_gemm1_wmma(const float* __restrict__ x, const float* __restrict__ W1,
               float* __restrict__ h1, int N) {
  int gtid = blockIdx.x * blockDim.x + threadIdx.x;
  int wave = gtid >> 5;
  int lane = threadIdx.x & 31;
  int half = lane >> 4;       // 0: K=0,1   1: K=2,3
  int m    = lane & 15;       // node-in-tile for A, feature for B/D
  int node0 = wave * 16;
  if (node0 >= N) return;     // uniform per wave: EXEC all-ones for WMMA

  int node = node0 + m;
  int nc   = node < N ? node : N - 1;   // clamp (tail stores guarded below)

  // Unconditional loads, value-select for the K=3 zero pad.
  float xa = x[(size_t)nc * 3 + (half ? 2 : 0)];  // K=0 or K=2
  float xb = x[(size_t)nc * 3 + 1];               // K=1 (unused K=3 -> 0)
  float wa = W1[(half ? 2 : 0) * 16 + m];
  float wb = W1[16 + m];

  v2f a, b;
  a.x = xa;
  a.y = half ? 0.0f : xb;
  b.x = wa;
  b.y = half ? 0.0f : wb;

  v8f c = {};
  c = __builtin_amdgcn_wmma_f32_16x16x4_f32(
      /*neg_a=*/false, a, /*neg_b=*/false, b,
      /*c_mod=*/(short)0, c, /*reuse_a=*/false, /*reuse_b=*/false);

  float* outBase = h1 + (size_t)(node0 + 8 * half) * 16 + m;
  if (node0 + 16 <= N) {
    // Full tile (always the case when N % 16 == 0): unguarded stores.
#pragma unroll
    for (int r = 0; r < 8; ++r) outBase[r * 16] = c[r];
  } else {
#pragma unroll
    for (int r = 0; r < 8; ++r) {
      int row = node0 + r + 8 * half;
      if (row < N) h1[(size_t)row * 16 + m] = c[r];
    }
  }
}

// ---------------------------------------------------------------------------
// 5) Layer-1 edge scatter: acc1[dst] += h1[src] * dinv[src]*dinv[dst]
//    One thread per (edge, feature): each wave covers 2 edges x 16 features,
//    so gathers and atomics are contiguous 64B rows per edge.
// ---------------------------------------------------------------------------
__global__ void gcn_scatter1(const int* __restrict__ src,
                             const int* __restrict__ dst,
                             const float* __restrict__ dinv,
                             const float* __restrict__ h1,
                             float* __restrict__ acc1, int E) {
  int t = blockIdx.x * blockDim.x + threadIdx.x;
  int e = t >> 4;
  int f = t & 15;
  if (e >= E) return;
  int s = src[e];
  int d = dst[e];
  float nrm = dinv[s] * dinv[d];
  float msg = h1[(size_t)s * 16 + f] * nrm;
  ATOMIC_ADD_F32(&acc1[(size_t)d * 16 + f], msg);
}

// ---------------------------------------------------------------------------
// 6) Per node: add self-loop msg + b1, relu, dot with W2 -> h2
// ---------------------------------------------------------------------------
__global__ void gcn_layer2_node(const float* __restrict__ acc1,
                                const float* __restrict__ h1,
                                const float* __restrict__ dinv,
                                const float* __restrict__ b1,
                                const float* __restrict__ W2,
                                float* __restrict__ h2, int N) {
  int i = blockIdx.x * blockDim.x + threadIdx.x;
  if (i >= N) return;
  float di = dinv[i];
  float sl = di * di;  // self-loop norm: dinv[i]*dinv[i]
  const float* ar = acc1 + (size_t)i * 16;
  const float* hr = h1 + (size_t)i * 16;
  float dot = 0.0f;
#pragma unroll
  for (int f = 0; f < 16; ++f) {
    float a = ar[f] + hr[f] * sl + b1[f];
    a = fmaxf(a, 0.0f);          // relu
    dot = fmaf(a, W2[f], dot);   // a1 @ W2  (16 -> 1)
  }
  h2[i] = dot;
}

// ---------------------------------------------------------------------------
// 7) Layer-2 edge scatter: acc2[dst] += h2[src] * norm
// ---------------------------------------------------------------------------
__global__ void gcn_scatter2(const int* __restrict__ src,
                             const int* __restrict__ dst,
                             const float* __restrict__ dinv,
                             const float* __restrict__ h2,
                             float* __restrict__ acc2, int E) {
  int e = blockIdx.x * blockDim.x + threadIdx.x;
  if (e >= E) return;
  int s = src[e];
  int d = dst[e];
  ATOMIC_ADD_F32(&acc2[d], h2[s] * dinv[s] * dinv[d]);
}

// ---------------------------------------------------------------------------
// 8) out = sigmoid(acc2 + self-loop + b2)
// ---------------------------------------------------------------------------
__global__ void gcn_final(const float* __restrict__ acc2,
                          const float* __restrict__ h2,
                          const float* __restrict__ dinv,
                          const float* __restrict__ b2,
                          float* __restrict__ out, int N) {
  int i = blockIdx.x * blockDim.x + threadIdx.x;
  if (i >= N) return;
  float di = dinv[i];
  float v = acc2[i] + h2[i] * di * di + b2[0];
  out[i] = 1.0f / (1.0f + __expf(-v));
}

// ---------------------------------------------------------------------------
// Launch
// ---------------------------------------------------------------------------
static inline size_t ws_align(size_t x) { return (x + 255) & ~(size_t)255; }

extern "C" void kernel_launch(void* const* d_in, const int* in_sizes, int n_in,
                              void* d_out, int out_size, void* d_ws, size_t ws_size,
                              hipStream_t stream) {
  const float* x  = (const float*)d_in[0];   // [N,3]
  const int*   ei = (const int*)d_in[1];     // [2,E] flattened (int32; JAX x64 off)
  const float* W1 = (const float*)d_in[2];   // [3,16]
  const float* b1 = (const float*)d_in[3];   // [16]
  const float* W2 = (const float*)d_in[4];   // [16,1]
  const float* b2 = (const float*)d_in[5];   // [1]
  float* out = (float*)d_out;                // [N,1]

  const int N = in_sizes[0] / 3;
  const int E = in_sizes[1] / 2;
  const int* src = ei;
  const int* dst = ei + E;

  // Workspace layout (256B-aligned slabs)
  char* ws = (char*)d_ws;
  size_t off = 0;
  float* deg  = (float*)(ws + off); off += ws_align((size_t)N * sizeof(float));       // -> dinv
  float* h1   = (float*)(ws + off); off += ws_align((size_t)N * 16 * sizeof(float));
  float* acc1 = (float*)(ws + off); off += ws_align((size_t)N * 16 * sizeof(float));
  float* h2   = (float*)(ws + off); off += ws_align((size_t)N * sizeof(float));
  float* acc2 = (float*)(ws + off); off += ws_align((size_t)N * sizeof(float));
  (void)ws_size;

  const int BLK = 256;
  int gInit  = (N * 16 + BLK - 1) / BLK;
  int gE     = (E + BLK - 1) / BLK;
  int gEF    = (int)(((long long)E * 16 + BLK - 1) / BLK);  // edge*feature threads
  int gN     = (N + BLK - 1) / BLK;
  int waves  = (N + 15) / 16;                 // one wave per 16-node tile
  int gGemm  = (waves * 32 + BLK - 1) / BLK;

  gcn_init<<<gInit, BLK, 0, stream>>>(deg, acc1, acc2, N);
  gcn_degree<<<gE, BLK, 0, stream>>>(dst, deg, E);
  gcn_dinv<<<gN, BLK, 0, stream>>>(deg, N);
  gcn_gemm1_wmma<<<gGemm, BLK, 0, stream>>>(x, W1, h1, N);
  gcn_scatter1<<<gEF, BLK, 0, stream>>>(src, dst, deg, h1, acc1, E);
  gcn_layer2_node<<<gN, BLK, 0, stream>>>(acc1, h1, deg, b1, W2, h2, N);
  gcn_scatter2<<<gE, BLK, 0, stream>>>(src, dst, deg, h2, acc2, E);
  gcn_final<<<gN, BLK, 0, stream>>>(acc2, h2, deg, b2, out, N);
}